// WindowAttention_70609262346702
// MI455X (gfx1250) — compile-verified
//
#include <hip/hip_runtime.h>
#include <hip/hip_bf16.h>

typedef float v2f __attribute__((ext_vector_type(2)));
typedef float v8f __attribute__((ext_vector_type(8)));

#define BB 2048
#define NN 64
#define CC 512
#define HH 8
#define DD 64
#define LS 68   // LDS row stride (floats): 68*4 = 272 B keeps every row 16B-aligned, avoids stride-64 bank conflicts

__global__ __launch_bounds__(128)
void win_attn_f32_wmma(const float* __restrict__ q1,
                       const float* __restrict__ q2,
                       const float* __restrict__ k1,
                       const float* __restrict__ v1,
                       const float* __restrict__ bias_table,
                       const float* __restrict__ gating,
                       const int*   __restrict__ rel_index,
                       float*       __restrict__ out)
{
    __shared__ float q1s[NN * LS];
    __shared__ float q2s[NN * LS];
    __shared__ float k1s[NN * LS];
    __shared__ float v1s[NN * LS];
    __shared__ float bps[NN * LS];   // bias, later reused as P (attn) staging per wave

    const int bh   = blockIdx.x;
    const int b    = bh >> 3;        // HH == 8
    const int h    = bh & 7;
    const int tid  = threadIdx.x;
    const int lane = tid & 31;
    const int wv   = tid >> 5;       // wave id 0..3, owns rows [wv*16, wv*16+16)
    const int hf   = lane >> 4;      // half-wave: 0 or 1
    const int l16  = lane & 15;

    const size_t base  = (size_t)b * (NN * CC) + (size_t)h * DD;
    const float  scale = 0.125f;     // 64^-0.5

    // ---------------- Stage inputs into LDS (float4, coalesced) ----------------
    // 64 rows x 16 float4 chunks per tensor; 128 threads -> 8 iterations each
    for (int c = tid; c < NN * 16; c += 128) {
        const int n  = c >> 4;
        const int d4 = (c & 15) << 2;
        const size_t g = base + (size_t)n * CC + d4;
        float4 a = *(const float4*)(q1 + g);
        float4 qb = *(const float4*)(q2 + g);
        float4 kk = *(const float4*)(k1 + g);
        float4 vv = *(const float4*)(v1 + g);
        a.x *= scale; a.y *= scale; a.z *= scale; a.w *= scale;
        qb.x *= scale; qb.y *= scale; qb.z *= scale; qb.w *= scale;
        *(float4*)(q1s + n * LS + d4) = a;
        *(float4*)(q2s + n * LS + d4) = qb;
        *(float4*)(k1s + n * LS + d4) = kk;
        *(float4*)(v1s + n * LS + d4) = vv;
    }
    // Relative-position bias gather: bias[n][m] = bias_table[rel_index[n*64+m]*H + h]
    for (int i = tid; i < NN * NN; i += 128) {
        const int n = i >> 6;
        const int m = i & 63;
        bps[n * LS + m] = bias_table[rel_index[i] * HH + h];
    }
    __syncthreads();

    // ---------------- S1 = Q1*K1^T + bias, S2 = Q2*K1^T + bias (WMMA f32) ------
    // C/D 16x16 f32 layout: VGPR v -> row (v + 8*hf) of the tile, col = l16.
    v8f c1[4], c2[4];
#pragma unroll
    for (int t = 0; t < 4; ++t) {
#pragma unroll
        for (int v = 0; v < 8; ++v) {
            const float bv = bps[(wv * 16 + v + hf * 8) * LS + t * 16 + l16];
            c1[t][v] = bv;
            c2[t][v] = bv;
        }
    }

    const int arow = wv * 16 + l16;      // A 16x4 layout: lanes0-15 M=lane K={k,k+1}; lanes16-31 M=lane-16 K={k+2,k+3}
#pragma unroll
    for (int k = 0; k < DD; k += 4) {
        const int kc = k + hf * 2;
        const v2f a1 = *(const v2f*)(q1s + arow * LS + kc);
        const v2f a2 = *(const v2f*)(q2s + arow * LS + kc);
#pragma unroll
        for (int t = 0; t < 4; ++t) {
            // B 4x16 layout: VGPR r -> K=k+r (lanes0-15) / K=k+r+2 (lanes16-31), N=l16.
            // B[K][n] = k1[n][K] -> contiguous float2 in LDS.
            const v2f bf = *(const v2f*)(k1s + (t * 16 + l16) * LS + kc);
            c1[t] = __builtin_amdgcn_wmma_f32_16x16x4_f32(false, a1, false, bf,
                                                          (short)0, c1[t], false, false);
            c2[t] = __builtin_amdgcn_wmma_f32_16x16x4_f32(false, a2, false, bf,
                                                          (short)0, c2[t], false, false);
        }
    }

    // ---------------- dual softmax + sigmoid gate (blend sums to 1 -> skip renorm) ----
    const float gp = gating[h];
    const float g  = 1.0f / (1.0f + __expf(-gp));
    const float og = 1.0f - g;

#pragma unroll
    for (int v = 0; v < 8; ++v) {
        float m1 = fmaxf(fmaxf(c1[0][v], c1[1][v]), fmaxf(c1[2][v], c1[3][v]));
        float m2 = fmaxf(fmaxf(c2[0][v], c2[1][v]), fmaxf(c2[2][v], c2[3][v]));
#pragma unroll
        for (int m = 1; m <= 8; m <<= 1) {       // butterfly stays within 16-lane halves
            m1 = fmaxf(m1, __shfl_xor(m1, m, 32));
            m2 = fmaxf(m2, __shfl_xor(m2, m, 32));
        }
        float e1[4], e2[4];
        float s1 = 0.0f, s2 = 0.0f;
#pragma unroll
        for (int t = 0; t < 4; ++t) {
            e1[t] = __expf(c1[t][v] - m1);
            e2[t] = __expf(c2[t][v] - m2);
            s1 += e1[t];
            s2 += e2[t];
        }
#pragma unroll
        for (int m = 1; m <= 8; m <<= 1) {
            s1 += __shfl_xor(s1, m, 32);
            s2 += __shfl_xor(s2, m, 32);
        }
        const float r1 = og / s1;
        const float r2 = g / s2;
#pragma unroll
        for (int t = 0; t < 4; ++t)
            c1[t][v] = e1[t] * r1 + e2[t] * r2;   // blended attention P
    }

    // ---------------- restage P (C-layout -> A-layout) via this wave's LDS strip ------
    // Reuse bps rows [wv*16, wv*16+16): same wave read bias there earlier; DS ops are
    // in-order within a wave so the WAR is safe. No cross-wave sharing.
    float* myp = bps + wv * 16 * LS;
#pragma unroll
    for (int t = 0; t < 4; ++t)
#pragma unroll
        for (int v = 0; v < 8; ++v)
            myp[(v + hf * 8) * LS + t * 16 + l16] = c1[t][v];

    // ---------------- O = P * V (WMMA f32) --------------------------------------------
    v8f acc[4];
#pragma unroll
    for (int t = 0; t < 4; ++t)
#pragma unroll
        for (int v = 0; v < 8; ++v)
            acc[t][v] = 0.0f;

#pragma unroll
    for (int k = 0; k < NN; k += 4) {
        const int kc = k + hf * 2;
        const v2f pa = *(const v2f*)(myp + l16 * LS + kc);   // P rows as A-fragments
#pragma unroll
        for (int t = 0; t < 4; ++t) {
            v2f pb;                                          // B[K][d] = v1[K][d], row-major
            pb[0] = v1s[(kc + 0) * LS + t * 16 + l16];
            pb[1] = v1s[(kc + 1) * LS + t * 16 + l16];
            acc[t] = __builtin_amdgcn_wmma_f32_16x16x4_f32(false, pa, false, pb,
                                                           (short)0, acc[t], false, false);
        }
    }

    // ---------------- write output (B, N, C) ------------------------------------------
    float* op = out + base;
#pragma unroll
    for (int t = 0; t < 4; ++t)
#pragma unroll
        for (int v = 0; v < 8; ++v) {
            const int n = wv * 16 + v + hf * 8;
            const int d = t * 16 + l16;
            op[(size_t)n * CC + d] = acc[t][v];
        }
}

extern "C" void kernel_launch(void* const* d_in, const int* in_sizes, int n_in,
                              void* d_out, int out_size, void* d_ws, size_t ws_size,
                              hipStream_t stream) {
    const float* q1 = (const float*)d_in[0];
    const float* q2 = (const float*)d_in[1];
    const float* k1 = (const float*)d_in[2];
    const float* v1 = (const float*)d_in[3];
    // d_in[4] = k2: unused by the reference math
    const float* bt = (const float*)d_in[5];
    const float* gp = (const float*)d_in[6];
    const int*   ri = (const int*)d_in[7];
    float* out = (float*)d_out;

    dim3 grid(BB * HH);
    dim3 block(128);
    hipLaunchKernelGGL(win_attn_f32_wmma, grid, block, 0, stream,
                       q1, q2, k1, v1, bt, gp, ri, out);
}